// DWT1DForward_66400194396237
// MI455X (gfx1250) — compile-verified
//
#include <hip/hip_runtime.h>
#include <cstdint>

// ---------------------------------------------------------------------------
// 3-level fused periodic db4 DWT, (16,64,65536) fp32.
// Memory-bound (3.5 FLOP/byte -> ~22us HBM floor at 23.3 TB/s): one HBM pass
// via CDNA5 async global->LDS DMA (GLOBAL_LOAD_ASYNC_TO_LDS_B128 / ASYNCcnt),
// all 3 wavelet levels fused in LDS (saves ~384 MB of intermediate traffic),
// non-temporal hints on both streams so they don't thrash the 192 MB L2.
// ---------------------------------------------------------------------------

#define NROW   65536
#define NMASK  65535
#define T3     1024                 // level-3 outputs per workgroup
#define THREADS 256
#define XLEN   (8*T3 + 56)          // staged input samples (incl. halo, padded)
#define L1LEN  (4*T3 + 21)          // lo1 halo window
#define L2LEN  (2*T3 + 7)           // lo2 halo window

// flat output layout: (x0, high1, high2, high3), R = 16*64 = 1024 rows
#define O_HI1  ((size_t)8388608)    // R*8192
#define O_HI2  ((size_t)41943040)   // + R*32768
#define O_HI3  ((size_t)58720256)   // + R*16384

__global__ __launch_bounds__(THREADS)
void dwt_db4_j3_kernel(const float* __restrict__ x,
                       const float* __restrict__ h0,
                       const float* __restrict__ h1,
                       float* __restrict__ out)
{
    __shared__ __align__(16) float sx  [XLEN];
    __shared__ __align__(16) float slo1[4*T3 + 24];
    __shared__ __align__(16) float slo2[2*T3 + 8];

    const int bid = blockIdx.x;
    const int row = bid >> 3;              // 8 tiles per row
    const int s3  = (bid & 7) * T3;        // tile's first level-3 output index
    const int tid = threadIdx.x;

    const float* xrow = x + (size_t)row * NROW;   // uniform -> SGPR pair

    // ---- async stage: sx[i] = x[(XB + i) mod N], XB = 8*s3 - 24 ------------
    // chunk starts are multiples of 4 floats (16B) -> no chunk straddles the
    // periodic wrap; 16B-aligned on both global and LDS sides. NT hint: each
    // input element is consumed by exactly one workgroup (streaming).
    const int XB = (8*s3 - 24 + NROW) & NMASK;
    const uint32_t sx_base = (uint32_t)(uintptr_t)(void*)sx;  // LDS byte addr
    const int NCHUNK = XLEN / 4;                              // 2062 x 16B
    for (int c = tid; c < NCHUNK; c += THREADS) {
        uint32_t goff = (uint32_t)(((XB + 4*c) & NMASK) * 4); // byte off in row
        uint32_t lptr = sx_base + 16u * (uint32_t)c;
        asm volatile("global_load_async_to_lds_b128 %0, %1, %2 th:TH_LOAD_NT"
                     :: "v"(lptr), "v"(goff), "s"(xrow) : "memory");
    }

    // filter taps (8 lo + 8 hi) -> registers while the DMA runs
    float f0[8], f1[8];
#pragma unroll
    for (int t = 0; t < 8; ++t) { f0[t] = h0[t]; f1[t] = h1[t]; }

    asm volatile("s_wait_asynccnt 0" ::: "memory");
    __syncthreads();

    // ---- level 1: lo1 halo into LDS, hi1 straight to HBM -------------------
    // stride-2 fp32 LDS reads across 32 lanes hit 32 distinct banks: no
    // conflicts.
    for (int m = tid; m < L1LEN; m += THREADS) {
        float a = 0.f;
#pragma unroll
        for (int t = 0; t < 8; ++t) a += f0[t] * sx[2*m + 3 + t];
        slo1[m] = a;
    }
    {
        float* o1 = out + O_HI1 + (size_t)row * 32768 + (size_t)(4*s3);
        for (int m = tid; m < 4*T3; m += THREADS) {
            float a = 0.f;
#pragma unroll
            for (int t = 0; t < 8; ++t) a += f1[t] * sx[2*m + 21 + t];
            __builtin_nontemporal_store(a, &o1[m]);
        }
    }
    __syncthreads();

    // ---- level 2 -----------------------------------------------------------
    for (int m = tid; m < L2LEN; m += THREADS) {
        float a = 0.f;
#pragma unroll
        for (int t = 0; t < 8; ++t) a += f0[t] * slo1[2*m + t];
        slo2[m] = a;
    }
    {
        float* o2 = out + O_HI2 + (size_t)row * 16384 + (size_t)(2*s3);
        for (int m = tid; m < 2*T3; m += THREADS) {
            float a = 0.f;
#pragma unroll
            for (int t = 0; t < 8; ++t) a += f1[t] * slo1[2*m + 6 + t];
            __builtin_nontemporal_store(a, &o2[m]);
        }
    }
    __syncthreads();

    // ---- level 3: lo3 (= x0) and hi3 ---------------------------------------
    {
        float* o0 = out +          (size_t)row * 8192 + (size_t)s3;
        float* o3 = out + O_HI3 + (size_t)row * 8192 + (size_t)s3;
        for (int m = tid; m < T3; m += THREADS) {
            float lo = 0.f, hi = 0.f;
#pragma unroll
            for (int t = 0; t < 8; ++t) {
                float v = slo2[2*m + t];
                lo += f0[t] * v;
                hi += f1[t] * v;
            }
            __builtin_nontemporal_store(lo, &o0[m]);
            __builtin_nontemporal_store(hi, &o3[m]);
        }
    }
}

extern "C" void kernel_launch(void* const* d_in, const int* in_sizes, int n_in,
                              void* d_out, int out_size, void* d_ws, size_t ws_size,
                              hipStream_t stream) {
    (void)in_sizes; (void)n_in; (void)out_size; (void)d_ws; (void)ws_size;
    const float* x  = (const float*)d_in[0];
    const float* h0 = (const float*)d_in[1];
    const float* h1 = (const float*)d_in[2];
    float* out = (float*)d_out;
    // 1024 rows * 8 tiles/row = 8192 workgroups of 256 threads (8 wave32s)
    hipLaunchKernelGGL(dwt_db4_j3_kernel, dim3(8192), dim3(THREADS), 0, stream,
                       x, h0, h1, out);
}